// DetectionCriterion_47674136985891
// MI455X (gfx1250) — compile-verified
//
#include <hip/hip_runtime.h>

// DetectionCriterion fused loss for MI455X (gfx1250, wave32).
// Memory-bound streaming reduction: ~524 MB reads -> one f32 scalar (~22us roofline @ 23.3 TB/s).
// Pass 1: per-batch-item pos/neg counts after hard-negative mining (cls+cmap = 105 MB,
//         fits in the 192 MB L2 so pass 2 re-reads them mostly from L2).
// Pass 2: fused class + reg loss with deterministic hash-based negative subsampling.
//         Key identity: class_loss == mine whenever cm_final != 0 (mining/balancing only
//         zeroes entries), so the second softplus of the reference is eliminated exactly.
// Wave reduction via V_WMMA_F32_16X16X4_F32: A(16x4) holds all 32 lanes' (p0,p1),
// B = ones -> D rows are row-sums; summing a lane's 8 acc VGPRs is uniform per half-wave,
// one SWAPX16 shuffle finishes the wave in full f32 precision.

#define N_T        25
#define IMG_HW     (128 * 128)
#define BATCH      32
#define CLS_PER_B  (N_T * IMG_HW)   // 409600
#define OUT_CH     (5 * N_T)        // 125
#define REG_CH     (4 * N_T)        // 100
#define HNM_THR    0.03f

typedef __attribute__((ext_vector_type(2))) float v2f;
typedef __attribute__((ext_vector_type(8))) float v8f;

__device__ __forceinline__ float softplusf(float x) {
    // stable softplus: max(x,0) + log(1 + exp(-|x|))
    return fmaxf(x, 0.0f) + __logf(1.0f + __expf(-fabsf(x)));
}

__device__ __forceinline__ float smooth_l1f(float x, float y) {
    float d = x - y;
    float ad = fabsf(d);
    return (ad < 1.0f) ? 0.5f * d * d : ad - 0.5f;
}

__device__ __forceinline__ float hash01(unsigned int x) {
    // deterministic per-element uniform [0,1) stand-in for jax.random.uniform subsample
    x ^= x >> 17; x *= 0xed5ad4bbu;
    x ^= x >> 11; x *= 0xac4c1b51u;
    x ^= x >> 15; x *= 0x31848babu;
    x ^= x >> 14;
    return (float)(x >> 8) * (1.0f / 16777216.0f);
}

__global__ void dc_init_kernel(float* __restrict__ result, unsigned int* __restrict__ counts) {
    if (threadIdx.x == 0) result[0] = 0.0f;
    if (threadIdx.x < 2 * BATCH) counts[threadIdx.x] = 0u;
}

// Pass 1: count positives / negatives per batch item after mining.
__global__ void dc_count_kernel(const float* __restrict__ out,
                                const float* __restrict__ cmap,
                                unsigned int* __restrict__ counts) {
    __shared__ unsigned int s_pos, s_neg;
    if (threadIdx.x == 0) { s_pos = 0u; s_neg = 0u; }
    __syncthreads();

    const int b = blockIdx.y;
    const float4* __restrict__ cls4 = (const float4*)(out + (size_t)b * OUT_CH * IMG_HW);
    const float4* __restrict__ cm4  = (const float4*)(cmap + (size_t)b * CLS_PER_B);
    const int nvec = CLS_PER_B / 4;

    unsigned int lp = 0u, ln = 0u;
    for (int i = blockIdx.x * blockDim.x + threadIdx.x; i < nvec;
         i += gridDim.x * blockDim.x) {
        float4 c = cls4[i];
        float4 m = cm4[i];
        float cl[4] = {c.x, c.y, c.z, c.w};
        float cm[4] = {m.x, m.y, m.z, m.w};
#pragma unroll
        for (int k = 0; k < 4; ++k) {
            float mine = softplusf(-cm[k] * cl[k]);
            float cmm = (mine < HNM_THR) ? 0.0f : cm[k];
            lp += (cmm > 0.0f) ? 1u : 0u;
            ln += (cmm < 0.0f) ? 1u : 0u;
        }
    }
    atomicAdd(&s_pos, lp);
    atomicAdd(&s_neg, ln);
    __syncthreads();
    if (threadIdx.x == 0) {
        atomicAdd(&counts[2 * b + 0], s_pos);
        atomicAdd(&counts[2 * b + 1], s_neg);
    }
}

// Pass 2: fused class + regression loss -> scalar.
__global__ void dc_loss_kernel(const float* __restrict__ out,
                               const float* __restrict__ cmap,
                               const float* __restrict__ rmap,
                               const unsigned int* __restrict__ counts,
                               float* __restrict__ result) {
    __shared__ float s_wave[8];

    const int b = blockIdx.y;
    const unsigned int npos = counts[2 * b + 0];
    const unsigned int nneg = counts[2 * b + 1];
    // POS_FRACTION = 0.5 -> n_keep = min(n_neg, n_pos)
    const unsigned int nkeep = (nneg < npos) ? nneg : npos;
    const float keep_ratio = (nneg > 0u) ? ((float)nkeep / (float)nneg) : 0.0f;

    const size_t out_base = (size_t)b * OUT_CH * IMG_HW;
    const size_t reg_base = (size_t)b * REG_CH * IMG_HW;
    const float4* __restrict__ cls4 = (const float4*)(out + out_base);
    const float4* __restrict__ cm4  = (const float4*)(cmap + (size_t)b * CLS_PER_B);
    const int nvec = CLS_PER_B / 4;

    float p0 = 0.0f;  // class loss partial
    float p1 = 0.0f;  // reg loss partial

    for (int i = blockIdx.x * blockDim.x + threadIdx.x; i < nvec;
         i += gridDim.x * blockDim.x) {
        float4 c = cls4[i];
        float4 m = cm4[i];
        float cl[4] = {c.x, c.y, c.z, c.w};
        float cm[4] = {m.x, m.y, m.z, m.w};
        float pos_mask[4];
        bool any_pos = false;
#pragma unroll
        for (int k = 0; k < 4; ++k) {
            float mine = softplusf(-cm[k] * cl[k]);
            float cmm = (mine < HNM_THR) ? 0.0f : cm[k];
            if (cmm < 0.0f) {
                unsigned int e = (unsigned int)b * (unsigned int)CLS_PER_B
                               + (unsigned int)(i * 4 + k);
                if (hash01(e) >= keep_ratio) cmm = 0.0f;  // subsample negatives
            }
            // cm_final != 0 implies cm_final == cm, so class_loss == mine exactly.
            if (cmm != 0.0f) p0 += mine;
            pos_mask[k] = (cmm > 0.0f) ? 1.0f : 0.0f;
            any_pos = any_pos || (cmm > 0.0f);
        }
        if (any_pos) {
            // 4 consecutive pixels share (t, pix base); IMG_HW % 4 == 0
            const int t   = (i * 4) / IMG_HW;
            const int pix = (i * 4) % IMG_HW;
#pragma unroll
            for (int q = 0; q < 4; ++q) {
                const size_t ch = (size_t)(q * N_T + t) * IMG_HW + pix;
                float4 rv = *(const float4*)(out + out_base + (size_t)N_T * IMG_HW + ch);
                float4 gv = *(const float4*)(rmap + reg_base + ch);
                float rr[4] = {rv.x, rv.y, rv.z, rv.w};
                float gg[4] = {gv.x, gv.y, gv.z, gv.w};
#pragma unroll
                for (int k = 0; k < 4; ++k)
                    p1 += pos_mask[k] * smooth_l1f(rr[k], gg[k]);
            }
        }
    }

    // ---- wave32 reduction via V_WMMA_F32_16X16X4_F32 ----
    float s;
#if defined(__has_builtin) && __has_builtin(__builtin_amdgcn_wmma_f32_16x16x4_f32)
    {
        v2f a; a.x = p0; a.y = p1;
        v2f bb; bb.x = 1.0f; bb.y = 1.0f;
        v8f acc = {};
        acc = __builtin_amdgcn_wmma_f32_16x16x4_f32(
            /*neg_a=*/false, a, /*neg_b=*/false, bb,
            /*c_mod=*/(short)0, acc, /*reuse_a=*/false, /*reuse_b=*/false);
        s = acc[0] + acc[1] + acc[2] + acc[3] + acc[4] + acc[5] + acc[6] + acc[7];
        s += __shfl_xor(s, 16, 32);
    }
#else
    {
        s = p0 + p1;
#pragma unroll
        for (int off = 16; off > 0; off >>= 1) s += __shfl_xor(s, off, 32);
    }
#endif

    const int wave = threadIdx.x >> 5;
    const int lane = threadIdx.x & 31;
    if (lane == 0) s_wave[wave] = s;
    __syncthreads();
    if (threadIdx.x == 0) {
        float tot = 0.0f;
        const int nw = blockDim.x >> 5;
        for (int w = 0; w < nw; ++w) tot += s_wave[w];
        atomicAdd(result, tot);  // global_atomic_add_f32
    }
}

extern "C" void kernel_launch(void* const* d_in, const int* in_sizes, int n_in,
                              void* d_out, int out_size, void* d_ws, size_t ws_size,
                              hipStream_t stream) {
    (void)in_sizes; (void)n_in; (void)out_size; (void)ws_size;
    const float* out_t = (const float*)d_in[0];  // (32, 125, 128, 128) f32
    const float* cmap  = (const float*)d_in[1];  // (32,  25, 128, 128) f32
    const float* rmap  = (const float*)d_in[2];  // (32, 100, 128, 128) f32
    float* result = (float*)d_out;               // scalar f32
    unsigned int* counts = (unsigned int*)d_ws;  // 2 * BATCH uints

    dc_init_kernel<<<1, 64, 0, stream>>>(result, counts);
    dc_count_kernel<<<dim3(32, BATCH), 256, 0, stream>>>(out_t, cmap, counts);
    dc_loss_kernel<<<dim3(64, BATCH), 256, 0, stream>>>(out_t, cmap, rmap, counts, result);
}